// EnforcedNeuralODE_57518202028150
// MI455X (gfx1250) — compile-verified
//
#include <hip/hip_runtime.h>

// ---------------------------------------------------------------------------
// EnforcedNeuralODE on MI455X (gfx1250), fp32 WMMA (V_WMMA_F32_16X16X4_F32).
//
//   x_t = Wx x_{t-1} + Wf f_{t-1} + b,   Wx = fc_w[:, :64], Wf = fc_w[:, 64:]
//
// Chunked affine scan (L=64, C=64 chunks):
//   k_powers     : Q_j = Wx^j, j=1..64                      (1 block, serial 63)
//   k_copy_x0    : out[0] = inputs
//   k_chunk_scan : per (chunk, batch-slab) wave: z_j = Wx z_{j-1} + gamma_j,
//                  z_0 = 0, gamma fused from forcing; writes z into out[t]
//   k_boundaries : h_{c+1} = Q_64 h_c + z_64^{(c)} serially over chunks -> H[c]
//   k_fixup      : out[t] += Q_j H[c]  for t = c*64 + j  (fully parallel)
// ---------------------------------------------------------------------------

#define BATCH  256
#define STATE  64
#define FDIM   64
#define TSPAN  4096
#define NSTEPS (TSPAN - 1)        // 4095
#define CHUNK  64
#define NCHUNK 64                 // last chunk has 63 steps
#define NB     16                 // batch columns per wave
#define NSLAB  (BATCH / NB)       // 16

typedef __attribute__((ext_vector_type(2))) float v2f;
typedef __attribute__((ext_vector_type(8))) float v8f;

// D = A(16x4) * B(4x16) + C, fp32. Layouts (wave32):
//   A/B: elem(vgpr j, lane l) = row (l&15), k = j + 2*(l>>4)
//   C/D: elem(vgpr v, lane l) = M (v + 8*(l>>4)), N (l&15)
__device__ __forceinline__ v8f wmma4(v2f a, v2f b, v8f c) {
  return __builtin_amdgcn_wmma_f32_16x16x4_f32(
      false, a, false, b, (short)0, c, false, false);
}

__device__ __forceinline__ v8f pack8(float4 lo, float4 hi) {
  v8f r;
  r[0] = lo.x; r[1] = lo.y; r[2] = lo.z; r[3] = lo.w;
  r[4] = hi.x; r[5] = hi.y; r[6] = hi.z; r[7] = hi.w;
  return r;
}

// ---------------------------------------------------------------------------
// Kernel 0: Q_j = Wx^j for j = 1..64.  One block, 16 waves (4x4 tile grid).
// ---------------------------------------------------------------------------
__global__ __launch_bounds__(512) void k_powers(const float* __restrict__ fcw,
                                                float* __restrict__ Q) {
  __shared__ float buf[2][STATE * STATE];   // [m][k], double buffered
  const int tid  = threadIdx.x;
  const int lane = tid & 31;
  const int wave = tid >> 5;                // 0..15
  const int mt   = wave >> 2;               // M tile
  const int nt   = wave & 3;                // N tile
  const int h    = lane >> 4;               // lane half
  const int lo16 = lane & 15;

  // A = Wx rows for this M tile (resident).
  v2f a[16];
  const int m = 16 * mt + lo16;
#pragma unroll
  for (int kc = 0; kc < 16; ++kc) {
    const float* p = fcw + m * 128 + 4 * kc + 2 * h;
    v2f t = {p[0], p[1]};
    a[kc] = t;
  }

  // buf[0] = Wx  (also store Q_1).
  for (int i = tid; i < STATE * STATE; i += 512) {
    const int mm = i >> 6, kk = i & 63;
    const float v = fcw[mm * 128 + kk];
    buf[0][i] = v;
    Q[1 * STATE * STATE + i] = v;
  }
  __syncthreads();

  int cur = 0;
  for (int j = 2; j <= CHUNK; ++j) {
    v8f c = {};
#pragma unroll
    for (int kc = 0; kc < 16; ++kc) {
      const int k0 = 4 * kc + 2 * h;
      v2f b = {buf[cur][k0 * STATE + 16 * nt + lo16],
               buf[cur][(k0 + 1) * STATE + 16 * nt + lo16]};
      c = wmma4(a[kc], b, c);
    }
#pragma unroll
    for (int v = 0; v < 8; ++v) {
      const int mm = 16 * mt + v + 8 * h;
      const int nn = 16 * nt + lo16;
      buf[cur ^ 1][mm * STATE + nn] = c[v];
      Q[j * STATE * STATE + mm * STATE + nn] = c[v];
    }
    __syncthreads();
    cur ^= 1;
  }
}

// ---------------------------------------------------------------------------
// Kernel 1: out[0, :, :] = inputs
// ---------------------------------------------------------------------------
__global__ void k_copy_x0(const float* __restrict__ in, float* __restrict__ out) {
  const int i = blockIdx.x * blockDim.x + threadIdx.x;   // float4 index
  ((float4*)out)[i] = ((const float4*)in)[i];
}

// ---------------------------------------------------------------------------
// Kernel 2A: per (batch-slab, chunk) wave, local scan with gamma fused:
//   z_j = Wx z_{j-1} + Wf f_{t-1} + b,  z_0 = 0;  out[t] = z_j (pre-fixup)
// State kept transposed y[s, b] in LDS; Wx/Wf resident as WMMA A operands.
// ---------------------------------------------------------------------------
__global__ __launch_bounds__(32) void k_chunk_scan(
    const float* __restrict__ forcing, const float* __restrict__ fcw,
    const float* __restrict__ fcb, float* __restrict__ out) {
  __shared__ float y[STATE * NB];           // [s][nb]
  const int lane = threadIdx.x;
  const int slab = blockIdx.x;
  const int c    = blockIdx.y;
  const int b0   = slab * NB;
  const int h    = lane >> 4;
  const int n    = lane & 15;

  // Resident A operands: Wx and Wf (4 M-tiles x 16 K-chunks each).
  v2f awx[4][16], awf[4][16];
#pragma unroll
  for (int mt = 0; mt < 4; ++mt)
#pragma unroll
    for (int kc = 0; kc < 16; ++kc) {
      const float* p = fcw + (16 * mt + n) * 128 + 4 * kc + 2 * h;
      v2f tx = {p[0], p[1]};
      v2f tf = {p[64], p[65]};
      awx[mt][kc] = tx;
      awf[mt][kc] = tf;
    }

  // Bias in C/D layout.
  v8f cb[4];
#pragma unroll
  for (int mt = 0; mt < 4; ++mt)
#pragma unroll
    for (int v = 0; v < 8; ++v) cb[mt][v] = fcb[16 * mt + v + 8 * h];

  // z_0 = 0.
  for (int i = lane; i < STATE * NB; i += 32) y[i] = 0.f;

  const int nsteps = (c == NCHUNK - 1) ? (CHUNK - 1) : CHUNK;
  for (int j = 1; j <= nsteps; ++j) {
    const int t = c * CHUNK + j;
    const float* F = forcing + ((t - 1) * BATCH + b0 + n) * FDIM;
    v8f acc[4];
#pragma unroll
    for (int mt = 0; mt < 4; ++mt) acc[mt] = cb[mt];

#pragma unroll
    for (int kc = 0; kc < 16; ++kc) {
      const int k0 = 4 * kc + 2 * h;
      v2f fb = {F[k0], F[k0 + 1]};                       // B = f^T tile
      v2f zb = {y[k0 * NB + n], y[(k0 + 1) * NB + n]};   // B = z_{j-1} tile
#pragma unroll
      for (int mt = 0; mt < 4; ++mt) {
        acc[mt] = wmma4(awf[mt][kc], fb, acc[mt]);
        acc[mt] = wmma4(awx[mt][kc], zb, acc[mt]);
      }
    }

    float* o = out + (t * BATCH + b0 + n) * STATE;
#pragma unroll
    for (int mt = 0; mt < 4; ++mt) {
      const int sb = 16 * mt + 8 * h;
#pragma unroll
      for (int v = 0; v < 8; ++v) y[(sb + v) * NB + n] = acc[mt][v];
      float4 lo = {acc[mt][0], acc[mt][1], acc[mt][2], acc[mt][3]};
      float4 hi = {acc[mt][4], acc[mt][5], acc[mt][6], acc[mt][7]};
      *(float4*)(o + sb)     = lo;
      *(float4*)(o + sb + 4) = hi;
    }
  }
}

// ---------------------------------------------------------------------------
// Kernel 2B: chunk-boundary states, one wave per batch slab:
//   H[0] = inputs^T ; H[c] = Q_64 H[c-1] + z_64^{(c-1)}  (= x_{c*64})
// ---------------------------------------------------------------------------
__global__ __launch_bounds__(32) void k_boundaries(
    const float* __restrict__ x0, const float* __restrict__ Q,
    const float* __restrict__ zout, float* __restrict__ H) {
  __shared__ float y[STATE * NB];
  const int lane = threadIdx.x;
  const int slab = blockIdx.x;
  const int b0   = slab * NB;
  const int h    = lane >> 4;
  const int n    = lane & 15;

  v2f aql[4][16];
  const float* QL = Q + CHUNK * STATE * STATE;   // Q_64
#pragma unroll
  for (int mt = 0; mt < 4; ++mt)
#pragma unroll
    for (int kc = 0; kc < 16; ++kc) {
      const float* p = QL + (16 * mt + n) * STATE + 4 * kc + 2 * h;
      v2f t = {p[0], p[1]};
      aql[mt][kc] = t;
    }

  // h_0 = inputs^T into LDS and H[0].
#pragma unroll 4
  for (int it = 0; it < 32; ++it) {
    const int idx = it * 32 + lane;      // 0..1023
    const int s = idx >> 4, nn = idx & 15;
    const float v = x0[(b0 + nn) * STATE + s];
    y[s * NB + nn] = v;
    H[s * BATCH + b0 + nn] = v;
  }

  for (int cc = 1; cc < NCHUNK; ++cc) {
    const int t = cc * CHUNK;
    const float* zp = zout + (t * BATCH + b0 + n) * STATE;
    v8f acc[4];
#pragma unroll
    for (int mt = 0; mt < 4; ++mt) {
      const int sb = 16 * mt + 8 * h;
      acc[mt] = pack8(*(const float4*)(zp + sb), *(const float4*)(zp + sb + 4));
    }
#pragma unroll
    for (int kc = 0; kc < 16; ++kc) {
      const int k0 = 4 * kc + 2 * h;
      v2f b = {y[k0 * NB + n], y[(k0 + 1) * NB + n]};
#pragma unroll
      for (int mt = 0; mt < 4; ++mt) acc[mt] = wmma4(aql[mt][kc], b, acc[mt]);
    }
#pragma unroll
    for (int mt = 0; mt < 4; ++mt) {
      const int sb = 16 * mt + 8 * h;
#pragma unroll
      for (int v = 0; v < 8; ++v) {
        y[(sb + v) * NB + n] = acc[mt][v];
        H[(cc * STATE + sb + v) * BATCH + b0 + n] = acc[mt][v];
      }
    }
  }
}

// ---------------------------------------------------------------------------
// Kernel 2C: parallel fix-up  out[t] = z_t + Q_j H[c],  t = c*64 + j.
// ---------------------------------------------------------------------------
__global__ __launch_bounds__(32) void k_fixup(const float* __restrict__ Q,
                                              const float* __restrict__ H,
                                              float* __restrict__ out) {
  const int lane = threadIdx.x;
  const int slab = blockIdx.x;
  const int b0   = slab * NB;
  const int t    = (int)blockIdx.y + 1;            // 1..4095
  const int c    = (t - 1) >> 6;
  const int j    = t - c * CHUNK;                  // 1..64
  const int h    = lane >> 4;
  const int n    = lane & 15;

  const float* Qj = Q + j * STATE * STATE;
  v2f a[4][16];
#pragma unroll
  for (int mt = 0; mt < 4; ++mt)
#pragma unroll
    for (int kc = 0; kc < 16; ++kc) {
      const float* p = Qj + (16 * mt + n) * STATE + 4 * kc + 2 * h;
      v2f tt = {p[0], p[1]};
      a[mt][kc] = tt;
    }

  float* zp = out + (t * BATCH + b0 + n) * STATE;
  v8f acc[4];
#pragma unroll
  for (int mt = 0; mt < 4; ++mt) {
    const int sb = 16 * mt + 8 * h;
    acc[mt] = pack8(*(const float4*)(zp + sb), *(const float4*)(zp + sb + 4));
  }

  const float* Hc = H + c * STATE * BATCH;
#pragma unroll
  for (int kc = 0; kc < 16; ++kc) {
    const int k0 = 4 * kc + 2 * h;
    v2f b = {Hc[k0 * BATCH + b0 + n], Hc[(k0 + 1) * BATCH + b0 + n]};
#pragma unroll
    for (int mt = 0; mt < 4; ++mt) acc[mt] = wmma4(a[mt][kc], b, acc[mt]);
  }

#pragma unroll
  for (int mt = 0; mt < 4; ++mt) {
    const int sb = 16 * mt + 8 * h;
    float4 lo = {acc[mt][0], acc[mt][1], acc[mt][2], acc[mt][3]};
    float4 hi = {acc[mt][4], acc[mt][5], acc[mt][6], acc[mt][7]};
    *(float4*)(zp + sb)     = lo;
    *(float4*)(zp + sb + 4) = hi;
  }
}

// ---------------------------------------------------------------------------
extern "C" void kernel_launch(void* const* d_in, const int* in_sizes, int n_in,
                              void* d_out, int out_size, void* d_ws,
                              size_t ws_size, hipStream_t stream) {
  (void)in_sizes; (void)n_in; (void)out_size; (void)ws_size;
  const float* inputs  = (const float*)d_in[0];   // [256, 64]
  const float* forcing = (const float*)d_in[1];   // [4095, 256, 64]
  const float* fcw     = (const float*)d_in[2];   // [64, 128]
  const float* fcb     = (const float*)d_in[3];   // [64]
  float* out = (float*)d_out;                     // [4096, 256, 64]

  // Workspace: Q_j powers (65 * 64*64 fl) then boundary states H (64*64*256 fl).
  float* Q = (float*)d_ws;
  float* H = Q + (CHUNK + 1) * STATE * STATE;

  k_powers<<<1, 512, 0, stream>>>(fcw, Q);
  k_copy_x0<<<(BATCH * STATE / 4) / 256, 256, 0, stream>>>(inputs, out);
  k_chunk_scan<<<dim3(NSLAB, NCHUNK), 32, 0, stream>>>(forcing, fcw, fcb, out);
  k_boundaries<<<NSLAB, 32, 0, stream>>>(inputs, Q, out, H);
  k_fixup<<<dim3(NSLAB, NSTEPS), 32, 0, stream>>>(Q, H, out);
}